// MultiHeadAttention_EDGE_29961691857498
// MI455X (gfx1250) — compile-verified
//
#include <hip/hip_runtime.h>
#include <math.h>

typedef __attribute__((ext_vector_type(2))) float v2f;
typedef __attribute__((ext_vector_type(8))) float v8f;
typedef __attribute__((ext_vector_type(4))) unsigned int u32x4;
typedef __attribute__((ext_vector_type(8))) int i32x8;
typedef __attribute__((ext_vector_type(4))) int i32x4;

#define HEADS 8
#define DIM 128
#define KD 16
#define BATCH 4
#define SEQ 512
#define NTILES (SEQ / 16)
#define NORM 0.25f
#define PSTRIDE 532    // attn LDS row stride (dwords); 532 mod 64 = 20 -> conflict-free
#define ESTRIDE 132    // edge LDS row stride (dwords); TDM pads 4 dwords per 128

// LDS byte offset of a __shared__ object (generic -> AS3 -> int).
#define LDS_OFF(p) ((unsigned int)(size_t)(__attribute__((address_space(3))) void*)(p))

static __device__ __forceinline__ v8f wmma_f32(v2f a, v2f b, v8f c) {
  // V_WMMA_F32_16X16X4_F32 : D = A(16x4 f32) * B(4x16 f32) + C(16x16 f32)
  return __builtin_amdgcn_wmma_f32_16x16x4_f32(false, a, false, b, (short)0, c,
                                               false, false);
}

// ---------------------------------------------------------------------------
// TDM: load one 16x128 f32 tile (rows contiguous, stride 128) into LDS,
// padded to 132-dword LDS rows.  D# layout per CDNA5 ISA ch.8.
// ---------------------------------------------------------------------------
static __device__ __forceinline__ void tdm_load_tile(unsigned int lds_addr,
                                                     const float* gptr) {
  const unsigned long long ga = (unsigned long long)(size_t)gptr;
  u32x4 g0;
  g0.x = 1u;                              // count=1, user mode, no gather
  g0.y = lds_addr;                        // lds_addr (bytes)
  g0.z = (unsigned int)ga;                // global_addr[31:0]
  g0.w = (unsigned int)((ga >> 32) & 0x01FFFFFFu) | (2u << 30);  // addr[56:32] | type=2
  i32x8 g1;
  // mask=0 | data_size=2 (4B) | pad_enable | pad_interval=6 (512B) | pad_amount=3 (4 dw)
  g1.s0 = (2 << 16) | (1 << 20) | (6 << 22) | (3 << 25);
  g1.s1 = (int)(128u << 16);              // tensor_dim0[15:0]=128 (bits 79:48)
  g1.s2 = (int)(16u << 16);               // tensor_dim1[15:0]=16  (bits 111:80)
  g1.s3 = (int)(128u << 16);              // tile_dim0=128         (bits 127:112)
  g1.s4 = 16;                             // tile_dim1=16, tile_dim2=0
  g1.s5 = 128;                            // tensor_dim0_stride=128
  g1.s6 = 0;                              // stride hi / dim1_stride lo
  g1.s7 = 0;
  const i32x4 z4 = {0, 0, 0, 0};
  const i32x8 z8 = {0, 0, 0, 0, 0, 0, 0, 0};
  // clang-23 / therock-10.0 6-arg form: (g0, g1, g2, g3, g4, cpol)
  __builtin_amdgcn_tensor_load_to_lds(g0, g1, z4, z4, z8, 0);
}

// ---------------------------------------------------------------------------
// Kernel 1: Q/K/V projections. One wave per (b, h, 16-row tile of n).
// Q is pre-scaled by norm = 1/sqrt(key_dim).
// ---------------------------------------------------------------------------
__global__ __launch_bounds__(32) void qkv_proj_kernel(
    const float* __restrict__ q, const float* __restrict__ Wq,
    const float* __restrict__ Wk, const float* __restrict__ Wv,
    float* __restrict__ Qs, float* __restrict__ Ks, float* __restrict__ Vs) {
  const int lane = threadIdx.x;
  const int l2 = lane & 15, hi = lane >> 4;
  const int tile = blockIdx.x % NTILES;
  const int bh = blockIdx.x / NTILES;
  const int h = bh % HEADS;
  const int b = bh / HEADS;
  const int n0 = tile * 16;

  const float* qrow = q + (size_t)(b * SEQ + n0 + l2) * DIM;
  const float* wq = Wq + (size_t)h * DIM * KD;
  const float* wk = Wk + (size_t)h * DIM * KD;
  const float* wv = Wv + (size_t)h * DIM * KD;

  v8f cq = {}, ck = {}, cv = {};
#pragma unroll
  for (int c = 0; c < 32; ++c) {
    const int k0 = 4 * c + 2 * hi;
    v2f a = *(const v2f*)(qrow + k0);
    v2f bq; bq.x = wq[k0 * KD + l2]; bq.y = wq[(k0 + 1) * KD + l2];
    v2f bk; bk.x = wk[k0 * KD + l2]; bk.y = wk[(k0 + 1) * KD + l2];
    v2f bv; bv.x = wv[k0 * KD + l2]; bv.y = wv[(k0 + 1) * KD + l2];
    cq = wmma_f32(a, bq, cq);
    ck = wmma_f32(a, bk, ck);
    cv = wmma_f32(a, bv, cv);
  }
  const size_t obase = ((size_t)(b * HEADS + h) * SEQ + n0) * KD;
#pragma unroll
  for (int j = 0; j < 8; ++j) {
    const int r = j + 8 * hi;
    Qs[obase + (size_t)r * KD + l2] = cq[j] * NORM;
    Ks[obase + (size_t)r * KD + l2] = ck[j];
    Vs[obase + (size_t)r * KD + l2] = cv[j];
  }
}

// ---------------------------------------------------------------------------
// Kernel 2: edge bias E[b,h,n,m] = dot(e[b,n,m,:], Ev[h,:]).
// 512 MB stream of e, fetched by the Tensor Data Mover into LDS with
// double buffering (TENSORcnt), A-operand fed from LDS, B held in VGPRs.
// 4 waves/block, each wave: 8 m-tiles of 16 rows.
// ---------------------------------------------------------------------------
__global__ __launch_bounds__(128) void edge_proj_kernel(
    const float* __restrict__ e, const float* __restrict__ Ev,
    float* __restrict__ Eb) {
  __shared__ float tile[4][2][16 * ESTRIDE];
  const int lane = threadIdx.x & 31;
  const int wave = __builtin_amdgcn_readfirstlane((int)(threadIdx.x >> 5));
  const int l2 = lane & 15, hi = lane >> 4;
  const int bn = blockIdx.x;  // b*SEQ + n
  const int n = bn & (SEQ - 1);
  const int b = bn >> 9;

  // Preload B = Ev^T (padded to 16 columns) into registers once.
  const float* evp = Ev + (size_t)(l2 < HEADS ? l2 : (HEADS - 1)) * DIM;
  v2f bb[32];
#pragma unroll
  for (int c = 0; c < 32; ++c) bb[c] = *(const v2f*)(evp + 4 * c + 2 * hi);

  const float* gbase = e + ((size_t)bn * SEQ + (size_t)wave * 8 * 16) * DIM;
  unsigned int lds0 = LDS_OFF(&tile[wave][0][0]);
  unsigned int lds1 = LDS_OFF(&tile[wave][1][0]);

  tdm_load_tile(lds0, gbase);
  asm volatile("" ::: "memory");

#pragma unroll 1
  for (int t = 0; t < 8; ++t) {
    if (t < 7) {
      // Buffer being refilled was last read in the previous iteration; make sure our
      // ds reads of it have drained before the TDM overwrites it.
      asm volatile("s_wait_dscnt 0x0" ::: "memory");
      tdm_load_tile((t & 1) ? lds0 : lds1, gbase + (size_t)(t + 1) * 16 * DIM);
      asm volatile("s_wait_tensorcnt 0x1" ::: "memory");  // tile t resident
    } else {
      asm volatile("s_wait_tensorcnt 0x0" ::: "memory");
    }
    const float* tl = &tile[wave][t & 1][0];
    v8f acc = {};
#pragma unroll
    for (int c = 0; c < 32; ++c) {
      const int k0 = 4 * c + 2 * hi;
      v2f a = *(const v2f*)(tl + l2 * ESTRIDE + k0);  // ds_load_b64, bank-clean
      acc = wmma_f32(a, bb[c], acc);
    }
    const int m0 = (wave * 8 + t) * 16;
    if (l2 < HEADS) {  // columns >= HEADS are padding; mask only the stores
      const size_t obase = ((size_t)(b * HEADS + l2) * SEQ + n) * SEQ + m0;
#pragma unroll
      for (int j = 0; j < 8; ++j) Eb[obase + j + 8 * hi] = acc[j];
    }
  }
}

// ---------------------------------------------------------------------------
// Kernel 3: fused attention per (b, h, 16-row n-tile):
//   S = Q K^T + E  -> LDS -> softmax rows -> P V  (K dim = 512)
// ---------------------------------------------------------------------------
__global__ __launch_bounds__(32) void attn_kernel(
    const float* __restrict__ Qs, const float* __restrict__ Ks,
    const float* __restrict__ Vs, const float* __restrict__ Eb,
    float* __restrict__ Hd) {
  __shared__ float P[16][PSTRIDE];
  __shared__ float invs[16];
  const int lane = threadIdx.x;
  const int l2 = lane & 15, hi = lane >> 4;
  const int tile = blockIdx.x % NTILES;
  const int bh = blockIdx.x / NTILES;  // b*HEADS + h
  const int n0 = tile * 16;

  const float* qrow = Qs + ((size_t)bh * SEQ + n0 + l2) * KD;
  v2f aq[4];
#pragma unroll
  for (int c = 0; c < 4; ++c) aq[c] = *(const v2f*)(qrow + 4 * c + 2 * hi);

  const float* kb = Ks + (size_t)bh * SEQ * KD;
  const float* ebase = Eb + ((size_t)bh * SEQ + n0) * SEQ;

  for (int m0 = 0; m0 < SEQ; m0 += 16) {
    v8f s = {};
#pragma unroll
    for (int c = 0; c < 4; ++c) {
      const int k0 = 4 * c + 2 * hi;
      v2f bk = *(const v2f*)(kb + (size_t)(m0 + l2) * KD + k0);  // B = K^T
      s = wmma_f32(aq[c], bk, s);
    }
#pragma unroll
    for (int j = 0; j < 8; ++j) {
      const int r = j + 8 * hi;
      P[r][m0 + l2] = s[j] + ebase[(size_t)r * SEQ + m0 + l2];
    }
  }
  __syncthreads();

#pragma unroll 1
  for (int r = 0; r < 16; ++r) {
    float mx = -__builtin_inff();
#pragma unroll
    for (int t = 0; t < 16; ++t) mx = fmaxf(mx, P[r][lane + 32 * t]);
#pragma unroll
    for (int off = 16; off > 0; off >>= 1)
      mx = fmaxf(mx, __shfl_xor(mx, off, 32));
    float sum = 0.f;
#pragma unroll
    for (int t = 0; t < 16; ++t) {
      const int idx = lane + 32 * t;
      const float v = __expf(P[r][idx] - mx);
      P[r][idx] = v;
      sum += v;
    }
#pragma unroll
    for (int off = 16; off > 0; off >>= 1) sum += __shfl_xor(sum, off, 32);
    if (lane == 0) invs[r] = 1.0f / sum;
  }
  __syncthreads();

  const float* vb = Vs + (size_t)bh * SEQ * KD;
  v8f acc = {};
  for (int c = 0; c < 128; ++c) {
    const int k0 = 4 * c + 2 * hi;
    v2f a = *(const v2f*)&P[l2][k0];
    v2f bv; bv.x = vb[(size_t)k0 * KD + l2]; bv.y = vb[(size_t)(k0 + 1) * KD + l2];
    acc = wmma_f32(a, bv, acc);
  }
  const size_t obase = ((size_t)bh * SEQ + n0) * KD;
#pragma unroll
  for (int j = 0; j < 8; ++j) {
    const int r = j + 8 * hi;
    Hd[obase + (size_t)r * KD + l2] = acc[j] * invs[r];
  }
}

// ---------------------------------------------------------------------------
// Kernel 4: out = heads[BN, H*16] x W_out[H*16, 128]  (W_out flat is [128,128])
// ---------------------------------------------------------------------------
__global__ __launch_bounds__(32) void outproj_kernel(
    const float* __restrict__ Hd, const float* __restrict__ Wo,
    float* __restrict__ out) {
  const int lane = threadIdx.x;
  const int l2 = lane & 15, hi = lane >> 4;
  const int et = blockIdx.x & 7;
  const int rest = blockIdx.x >> 3;
  const int nt = rest % NTILES;
  const int b = rest / NTILES;
  const int n0 = nt * 16, e0 = et * 16;

  const float* hb = Hd + (size_t)b * HEADS * SEQ * KD;
  v8f acc = {};
#pragma unroll
  for (int c = 0; c < 32; ++c) {
    const int k0 = 4 * c + 2 * hi;
    const int hk = k0 >> 4, vk = k0 & 15;
    v2f a = *(const v2f*)(hb + ((size_t)hk * SEQ + n0 + l2) * KD + vk);
    v2f bw; bw.x = Wo[(size_t)k0 * DIM + e0 + l2];
    bw.y = Wo[(size_t)(k0 + 1) * DIM + e0 + l2];
    acc = wmma_f32(a, bw, acc);
  }
#pragma unroll
  for (int j = 0; j < 8; ++j) {
    const int r = j + 8 * hi;
    out[((size_t)b * SEQ + n0 + r) * DIM + e0 + l2] = acc[j];
  }
}

// ---------------------------------------------------------------------------
extern "C" void kernel_launch(void* const* d_in, const int* in_sizes, int n_in,
                              void* d_out, int out_size, void* d_ws,
                              size_t ws_size, hipStream_t stream) {
  const float* q  = (const float*)d_in[0];   // [B,N,128]
  const float* e  = (const float*)d_in[1];   // [B,N,N,128]
  const float* Wq = (const float*)d_in[2];   // [H,128,16]
  const float* Wk = (const float*)d_in[3];
  const float* Wv = (const float*)d_in[4];
  const float* Ev = (const float*)d_in[5];   // [H,128,1]
  const float* Wo = (const float*)d_in[6];   // [H,16,128]
  float* out = (float*)d_out;                // [B,N,128]

  const size_t qkv = (size_t)BATCH * HEADS * SEQ * KD;
  float* ws = (float*)d_ws;
  float* Qs = ws;
  float* Ks = Qs + qkv;
  float* Vs = Ks + qkv;
  float* Hd = Vs + qkv;
  float* Eb = Hd + qkv;  // [B,H,N,N]

  qkv_proj_kernel<<<BATCH * HEADS * NTILES, 32, 0, stream>>>(q, Wq, Wk, Wv, Qs,
                                                             Ks, Vs);
  edge_proj_kernel<<<BATCH * SEQ, 128, 0, stream>>>(e, Ev, Eb);
  attn_kernel<<<BATCH * HEADS * NTILES, 32, 0, stream>>>(Qs, Ks, Vs, Eb, Hd);
  outproj_kernel<<<BATCH * NTILES * 8, 32, 0, stream>>>(Hd, Wo, out);
}